// MetapopLayer_27230092657388
// MI455X (gfx1250) — compile-verified
//
#include <hip/hip_runtime.h>
#include <hip/hip_bf16.h>

typedef float v2f __attribute__((ext_vector_type(2)));
typedef float v8f __attribute__((ext_vector_type(8)));

#define MM   64      // patches
#define CC   4       // compartments
#define BETA 0.25f

// One workgroup per sample `a`. 256 threads = 8 wave32.
// LDS: R[a] (row-padded), scrambled Rt[a], padded T[a], rho, scratch. ~36 KB.
__global__ __launch_bounds__(256)
void metapop_kernel(const float* __restrict__ R,
                    const float* __restrict__ T,
                    const float* __restrict__ rho0,
                    const int*   __restrict__ tsPtr,
                    float*       __restrict__ out,
                    int N)
{
    __shared__ float sR [MM * 65];      // sR[i*65 + j] = R[a,i,j]  (padded: conflict-free row dots)
    __shared__ float sRt[MM * MM];      // sRt[i*64 + j] = Rt[a,i,j] (lane<->j access, stride 64 ok)
    __shared__ float sTpad[CC * 16];    // B operand: T zero-padded to 4x16 (cols 4..15 = 0)
    __shared__ float sBinv[MM];         // BETA / ntot[j]
    __shared__ float sRho [MM * CC];
    __shared__ float sRhoN[MM * CC];
    __shared__ float sP  [MM];          // infect_prob
    __shared__ float sNI [MM];          // new_infected
    __shared__ float sPart[4 * MM];     // 4-segment reduction scratch

    const int a    = blockIdx.x;
    const int tid  = threadIdx.x;       // 0..255
    const int lane = tid & 31;
    const int wave = tid >> 5;

    // ---- stage R[a] (row-major, padded stride 65) ----
    const float* Ra = R + (size_t)a * MM * MM;
    for (int k = tid; k < MM * MM; k += 256) {
        int i = k >> 6, j = k & 63;
        sR[i * 65 + j] = Ra[k];
    }
    // ---- gather scrambled Rt[a]: faithful jnp.transpose(R).reshape(-1,M,M,1) ----
    // Rt[a,i,j] = R[f % N, (f/N) % M, f/(N*M)] with f = a*M*M + i*M + j
    for (int k = tid; k < MM * MM; k += 256) {
        int f = a * (MM * MM) + k;
        int z = f % N;
        int y = (f / N) % MM;
        int x = f / (N * MM);
        sRt[k] = R[((size_t)z * MM + y) * MM + x];
    }
    // ---- stage T[a] into zero-padded 4x16 B tile ----
    if (tid < CC * 16) {
        int k = tid >> 4, n = tid & 15;
        sTpad[tid] = (n < CC) ? T[(size_t)a * CC * CC + k * CC + n] : 0.0f;
    }
    sRho[tid] = rho0[(size_t)a * MM * CC + tid];   // 256 elements exactly
    __syncthreads();

    // ---- ntot[j] = sum_i R[a,i,j]; precompute BETA/ntot ----
    if (tid < MM) {
        float s = 0.f;
        for (int i = 0; i < MM; ++i) s += sR[i * 65 + tid];   // bank (i+j)&63: clean
        sBinv[tid] = BETA / s;
    }
    __syncthreads();

    const int TS = *tsPtr;
    const int j    = tid & 63;      // column / row index for dot phases
    const int seg  = tid >> 6;      // 4 segments of 16 (seg uniform per wave)
    const int ci   = tid >> 2;      // commit phase: patch index
    const int cc   = tid & 3;       // commit phase: compartment index
    const float isC0 = (cc == 0) ? 1.0f : 0.0f;

    for (int t = 0; t < TS; ++t) {
        // ---- Phase 1: S[j] = sum_i rho[i,1] * Rt[i,j] ----
        float acc = 0.f;
        for (int s = 0; s < 16; ++s) {
            int i = seg * 16 + s;                      // wave-uniform i
            acc += sRho[i * CC + 1] * sRt[i * MM + j]; // broadcast * consecutive
        }
        sPart[seg * MM + j] = acc;
        __syncthreads();
        if (tid < MM) {
            float S = sPart[tid] + sPart[MM + tid] + sPart[2 * MM + tid] + sPart[3 * MM + tid];
            sP[tid] = 1.0f - __expf(-S * sBinv[tid]);  // v_exp_f32 path
        }
        __syncthreads();

        // ---- Phase 2: ni[i] = (1 - sum_c rho[i,c]) * sum_j R[i,j]*p[j] ----
        float acc2 = 0.f;
        for (int s = 0; s < 16; ++s) {
            int jj = seg * 16 + s;                     // wave-uniform jj
            acc2 += sR[j * 65 + jj] * sP[jj];          // bank (i+jj)&63: clean
        }
        sPart[seg * MM + j] = acc2;
        __syncthreads();
        if (tid < MM) {
            float dot  = sPart[tid] + sPart[MM + tid] + sPart[2 * MM + tid] + sPart[3 * MM + tid];
            float frac = 1.0f - (sRho[tid * CC + 0] + sRho[tid * CC + 1]
                               + sRho[tid * CC + 2] + sRho[tid * CC + 3]);
            sNI[tid] = frac * dot;
        }
        __syncthreads();

        // ---- Phase 3: rho_new = rho @ T  via V_WMMA_F32_16X16X4_F32 ----
        // Waves 0..3 each own a 16-row tile (EXEC all-ones inside the wave).
        if (wave < 4) {
            const int ibase = wave * 16;
            const int m  = lane & 15;
            const int hi = lane >> 4;                  // 0: K=0,1  1: K=2,3
            v2f afrag, bfrag;
            // A (16x4 f32): VGPR0 = K0|K2 halves, VGPR1 = K1|K3 halves
            afrag[0] = sRho[(ibase + m) * CC + (hi ? 2 : 0)];
            afrag[1] = sRho[(ibase + m) * CC + (hi ? 3 : 1)];
            // B (4x16 f32): pre-padded tile -> unconditional loads
            bfrag[0] = sTpad[(hi ? 2 : 0) * 16 + m];
            bfrag[1] = sTpad[(hi ? 3 : 1) * 16 + m];
            v8f cz = {};
            v8f d = __builtin_amdgcn_wmma_f32_16x16x4_f32(
                        false, afrag, false, bfrag, (short)0, cz, false, false);
            // D layout: vgpr r -> rows r (lanes 0-15) / r+8 (lanes 16-31), col = lane&15
            if (m < CC) {                               // raw stores only; epilogue is full-width
                #pragma unroll
                for (int r = 0; r < 8; ++r) {
                    int row = ibase + r + (hi ? 8 : 0);
                    sRhoN[row * CC + m] = d[r];
                }
            }
        }
        __syncthreads();

        // ---- commit (full width, branch-free): e0 injection, clip, copy, stream ----
        float v = sRhoN[tid] + isC0 * sNI[ci];
        v = fminf(fmaxf(v, 0.0f), 1e10f);
        sRho[tid] = v;
        out[(((size_t)t * N + a) * (MM * CC)) + tid] = v;
        __syncthreads();
    }
}

extern "C" void kernel_launch(void* const* d_in, const int* in_sizes, int n_in,
                              void* d_out, int out_size, void* d_ws, size_t ws_size,
                              hipStream_t stream) {
    const float* R    = (const float*)d_in[0];
    const float* T    = (const float*)d_in[1];
    const float* rho0 = (const float*)d_in[2];
    const int*   ts   = (const int*)d_in[3];
    float* out = (float*)d_out;

    const int N = in_sizes[0] / (MM * MM);   // 1024 for reference shapes
    hipLaunchKernelGGL(metapop_kernel, dim3(N), dim3(256), 0, stream,
                       R, T, rho0, ts, out, N);
}